// SOM_Memory_21818433864238
// MI455X (gfx1250) — compile-verified
//
#include <hip/hip_runtime.h>

#define C_NUM 2048
#define D_DIM 8192
#define B_NUM 64

typedef __attribute__((ext_vector_type(2))) float v2f;
typedef __attribute__((ext_vector_type(8))) float v8f;

// -------------------------------------------------------------------------
// Kernel 1: G1[b][c] = dot(xf[b], centroid[c]); also x2[b], c2[c] norms.
// One wave per 16x16 output tile, f32 WMMA, K = 8192 in steps of 4.
// A fragment (16x4, MxK): lane<16 -> (M=lane, K=k..k+1), lane>=16 -> K=k+2..k+3
// B fragment (4x16, KxN): lane<16 -> (N=lane, K=k..k+1), lane>=16 -> K=k+2..k+3
// B[k][n] = centroid[n][k] => contiguous float2 per lane from a centroid row.
// -------------------------------------------------------------------------
__global__ __launch_bounds__(32)
void k_gemm1(const float* __restrict__ xf, const float* __restrict__ cen,
             float* __restrict__ G1, float* __restrict__ x2, float* __restrict__ c2)
{
    const int lane = threadIdx.x;
    const int sub  = lane & 15;
    const int h    = lane >> 4;          // which K-half of the fragment
    const int mt   = blockIdx.x & 3;     // 4 M-tiles (M=64)
    const int nt   = blockIdx.x >> 2;    // 128 N-tiles (N=2048)
    const int m0   = mt * 16, n0 = nt * 16;

    const float* arow = xf  + (size_t)(m0 + sub) * D_DIM + 2 * h;
    const float* brow = cen + (size_t)(n0 + sub) * D_DIM + 2 * h;

    v8f acc = {0.f, 0.f, 0.f, 0.f, 0.f, 0.f, 0.f, 0.f};
    float sa = 0.f, sb = 0.f;
    for (int k = 0; k < D_DIM; k += 4) {
        v2f a = *reinterpret_cast<const v2f*>(arow + k);
        v2f b = *reinterpret_cast<const v2f*>(brow + k);
        sa += a[0] * a[0] + a[1] * a[1];
        sb += b[0] * b[0] + b[1] * b[1];
        acc = __builtin_amdgcn_wmma_f32_16x16x4_f32(false, a, false, b,
                                                    (short)0, acc, false, false);
    }
#pragma unroll
    for (int r = 0; r < 8; ++r) {
        int row = m0 + r + 8 * h;                 // D-matrix VGPR layout
        G1[(size_t)row * C_NUM + n0 + sub] = acc[r];
    }
    // combine the two K-halves of the per-lane norm partials
    float oa = __shfl(sa, lane ^ 16, 32);
    float ob = __shfl(sb, lane ^ 16, 32);
    if (lane < 16) {
        if (nt == 0) x2[m0 + lane] = sa + oa;     // one writer set per row
        if (mt == 0) c2[n0 + lane] = sb + ob;     // one writer set per col
    }
}

// -------------------------------------------------------------------------
// Kernel 2: per-batch distance -> argmax (first index) + softmax -> prob, idx
// -------------------------------------------------------------------------
__global__ __launch_bounds__(256)
void k_softmax_argmax(const float* __restrict__ G1, const float* __restrict__ x2,
                      const float* __restrict__ c2, float* __restrict__ prob,
                      float* __restrict__ idx_f, int* __restrict__ idx_i)
{
    const int b = blockIdx.x;
    const int t = threadIdx.x;
    __shared__ float smax[256];
    __shared__ int   sidx[256];
    __shared__ float ssum[256];

    const float xb = x2[b];
    float dloc[8];
    float vmax = -3.4e38f;
    int   imax = 0;
#pragma unroll
    for (int j = 0; j < 8; ++j) {
        int c = t + j * 256;
        float g  = G1[(size_t)b * C_NUM + c];
        float sq = fmaxf(xb + c2[c] - 2.0f * g, 0.0f);
        float d  = -sqrtf(sq);
        dloc[j] = d;
        if (d > vmax) { vmax = d; imax = c; }     // strict > keeps first index
    }
    smax[t] = vmax; sidx[t] = imax;
    __syncthreads();
    for (int s = 128; s > 0; s >>= 1) {
        if (t < s) {
            float v = smax[t + s]; int i = sidx[t + s];
            if (v > smax[t] || (v == smax[t] && i < sidx[t])) { smax[t] = v; sidx[t] = i; }
        }
        __syncthreads();
    }
    const float gmax = smax[0];

    float e[8];
    float lsum = 0.f;
#pragma unroll
    for (int j = 0; j < 8; ++j) { e[j] = expf(dloc[j] - gmax); lsum += e[j]; }
    ssum[t] = lsum;
    __syncthreads();
    for (int s = 128; s > 0; s >>= 1) {
        if (t < s) ssum[t] += ssum[t + s];
        __syncthreads();
    }
    const float inv = 1.0f / ssum[0];
#pragma unroll
    for (int j = 0; j < 8; ++j)
        prob[(size_t)b * C_NUM + t + j * 256] = e[j] * inv;

    if (t == 0) { idx_f[b] = (float)sidx[0]; idx_i[b] = sidx[0]; }
}

// -------------------------------------------------------------------------
// Kernel 3: G2[b][c] = dot(centroid[idx[b]], centroid[c])  (bmu gather GEMM)
// -------------------------------------------------------------------------
__global__ __launch_bounds__(32)
void k_gemm2(const float* __restrict__ cen, const int* __restrict__ idx_i,
             float* __restrict__ G2)
{
    const int lane = threadIdx.x;
    const int sub  = lane & 15;
    const int h    = lane >> 4;
    const int mt   = blockIdx.x & 3;
    const int nt   = blockIdx.x >> 2;
    const int m0   = mt * 16, n0 = nt * 16;

    const int ar = idx_i[m0 + sub];   // same for lane pair (sub, sub+16)
    const float* arow = cen + (size_t)ar * D_DIM + 2 * h;
    const float* brow = cen + (size_t)(n0 + sub) * D_DIM + 2 * h;

    v8f acc = {0.f, 0.f, 0.f, 0.f, 0.f, 0.f, 0.f, 0.f};
    for (int k = 0; k < D_DIM; k += 4) {
        v2f a = *reinterpret_cast<const v2f*>(arow + k);
        v2f b = *reinterpret_cast<const v2f*>(brow + k);
        acc = __builtin_amdgcn_wmma_f32_16x16x4_f32(false, a, false, b,
                                                    (short)0, acc, false, false);
    }
#pragma unroll
    for (int r = 0; r < 8; ++r) {
        int row = m0 + r + 8 * h;
        G2[(size_t)row * C_NUM + n0 + sub] = acc[r];
    }
}

// -------------------------------------------------------------------------
// Kernel 4: w[b][c] = exp(-sqrt(clamp(b2+c2-2*G2,0)+1e-8)/200)
// stored transposed as wT[c][b]; colsum[c] = sum_b w[b][c]
// -------------------------------------------------------------------------
__global__ __launch_bounds__(256)
void k_weights(const float* __restrict__ G2, const float* __restrict__ c2,
               const int* __restrict__ idx_i, float* __restrict__ wT,
               float* __restrict__ colsum)
{
    __shared__ float b2s[B_NUM];
    const int t = threadIdx.x;
    if (t < B_NUM) b2s[t] = c2[idx_i[t]];
    __syncthreads();

    const int c = blockIdx.x * 256 + t;
    const float c2c = c2[c];
    float s = 0.f;
    for (int b = 0; b < B_NUM; ++b) {
        float g  = G2[(size_t)b * C_NUM + c];
        float sq = fmaxf(b2s[b] + c2c - 2.0f * g, 0.0f) + 1e-8f;
        float w  = expf(-sqrtf(sq) * 0.005f);     // /200
        s += w;
        wT[(size_t)c * B_NUM + b] = w;
    }
    colsum[c] = s;
}

// -------------------------------------------------------------------------
// Kernel 5: new_weights = centroid + (LR/B)*(wT' @ xf - colsum[:,None]*centroid)
// GEMM M=2048 (c), N=8192 (d), K=64 (b); one wave per 16x16 tile, fused epilogue.
// -------------------------------------------------------------------------
__global__ __launch_bounds__(256)
void k_update(const float* __restrict__ wT, const float* __restrict__ xf,
              const float* __restrict__ cen, const float* __restrict__ colsum,
              float* __restrict__ outw)
{
    const int gtid = blockIdx.x * 256 + threadIdx.x;
    const int wave = gtid >> 5;
    const int lane = gtid & 31;
    const int sub  = lane & 15;
    const int h    = lane >> 4;
    const int nt   = wave & 511;   // 512 N-tiles (D=8192)
    const int mt   = wave >> 9;    // 128 M-tiles (C=2048)
    const int m0   = mt * 16, n0 = nt * 16;

    const float* arow = wT + (size_t)(m0 + sub) * B_NUM + 2 * h;
    const float* bcol = xf + (n0 + sub);

    v8f acc = {0.f, 0.f, 0.f, 0.f, 0.f, 0.f, 0.f, 0.f};
#pragma unroll
    for (int k = 0; k < B_NUM; k += 4) {
        v2f a = *reinterpret_cast<const v2f*>(arow + k);
        v2f b;
        b[0] = bcol[(size_t)(k + 2 * h)     * D_DIM];   // B[k][n] = xf[k][n]
        b[1] = bcol[(size_t)(k + 2 * h + 1) * D_DIM];
        acc = __builtin_amdgcn_wmma_f32_16x16x4_f32(false, a, false, b,
                                                    (short)0, acc, false, false);
    }
    const float scale = 0.1f / (float)B_NUM;   // LR / B
#pragma unroll
    for (int r = 0; r < 8; ++r) {
        int row = m0 + r + 8 * h;
        size_t off = (size_t)row * D_DIM + n0 + sub;
        float cv = cen[off];
        outw[off] = cv + scale * (acc[r] - colsum[row] * cv);
    }
}

// -------------------------------------------------------------------------
extern "C" void kernel_launch(void* const* d_in, const int* in_sizes, int n_in,
                              void* d_out, int out_size, void* d_ws, size_t ws_size,
                              hipStream_t stream)
{
    const float* xf  = (const float*)d_in[0];   // [64, 64, 128] == [64, 8192]
    const float* cen = (const float*)d_in[1];   // [2048, 8192]

    float* prob  = (float*)d_out;                                   // [64, 2048]
    float* neww  = (float*)d_out + (size_t)B_NUM * C_NUM;           // [2048, 8192]
    float* idx_f = neww + (size_t)C_NUM * D_DIM;                    // [64]

    float* wsf    = (float*)d_ws;
    float* G1     = wsf;                               // 131072
    float* G2     = G1 + (size_t)B_NUM * C_NUM;        // 131072
    float* wT     = G2 + (size_t)B_NUM * C_NUM;        // 131072
    float* x2     = wT + (size_t)C_NUM * B_NUM;        // 64
    float* c2     = x2 + B_NUM;                        // 2048
    float* colsum = c2 + C_NUM;                        // 2048
    int*   idx_i  = (int*)(colsum + C_NUM);            // 64

    // 1) distance GEMM + norms: 512 tiles, one wave each
    k_gemm1<<<512, 32, 0, stream>>>(xf, cen, G1, x2, c2);
    // 2) argmax + softmax per batch row
    k_softmax_argmax<<<B_NUM, 256, 0, stream>>>(G1, x2, c2, prob, idx_f, idx_i);
    // 3) bmu-vs-centroid GEMM (gathered rows)
    k_gemm2<<<512, 32, 0, stream>>>(cen, idx_i, G2);
    // 4) neighborhood weights (transposed) + column sums
    k_weights<<<C_NUM / 256, 256, 0, stream>>>(G2, c2, idx_i, wT, colsum);
    // 5) update GEMM + fused codebook update: 65536 tiles, 8 waves/block
    k_update<<<(C_NUM / 16) * (D_DIM / 16) / 8, 256, 0, stream>>>(wT, xf, cen, colsum, neww);
}